// DynConv2d_5128190952114
// MI455X (gfx1250) — compile-verified
//
#include <hip/hip_runtime.h>

#define B_    8
#define N_    2048
#define C_    128
#define K_    9
#define C2_   256
#define COUT_ 256
#define BN_EPS 1e-5f

typedef float v2f __attribute__((ext_vector_type(2)));
typedef float v8f __attribute__((ext_vector_type(8)));
typedef unsigned int v4u __attribute__((ext_vector_type(4)));
typedef unsigned int v8u __attribute__((ext_vector_type(8)));

static __device__ __forceinline__ v8f wmma_f32_16x16x4(v2f a, v2f b, v8f c) {
  // D = A(16x4) * B(4x16) + C(16x16), all f32, wave32
  return __builtin_amdgcn_wmma_f32_16x16x4_f32(false, a, false, b, (short)0, c, false, false);
}

// --------------------------------------------------------------- TDM load
// 2-D tile load via Tensor Data Mover: tile_dim0=16 (row of 16 f32, 64B),
// tile_dim1=128 rows, row stride = tensor_dim0_stride = 2048 f32.
// Packed into LDS row-major [128][16] at lds_byte offset.
// D# group0/group1 per CDNA5 ISA 8.3/8.4; 2-group (<=2D) form of
// tensor_load_to_lds; tracked by TENSORcnt.
static __device__ __forceinline__ void tdm_load_tile16x128(unsigned lds_byte,
                                                           const void* gptr) {
  unsigned long long ga = (unsigned long long)(size_t)gptr;
  v4u g0;
  g0.x = 1u;                                   // count=1 (valid), user mode
  g0.y = lds_byte;                             // lds_addr (bytes)
  g0.z = (unsigned)ga;                         // global_addr[31:0]
  g0.w = (unsigned)((ga >> 32) & 0x01FFFFFFull) | (2u << 30); // addr[56:32]|type=2
  v8u g1;
  g1[0] = 2u << 16;        // workgroup_mask=0 (no cluster), data_size=2 (4B)
  g1[1] = 2048u << 16;     // tensor_dim0 = 2048 (bits[79:48], low half)
  g1[2] = 128u << 16;      // tensor_dim0 hi=0 | tensor_dim1 = 128
  g1[3] = 16u << 16;       // tensor_dim1 hi=0 | tile_dim0 = 16
  g1[4] = 128u;            // tile_dim1 = 128, tile_dim2 = 0
  g1[5] = 2048u;           // tensor_dim0_stride lo = 2048
  g1[6] = 0u;              // stride hi / dim1_stride lo (unused, 2D)
  g1[7] = 0u;
  asm volatile("tensor_load_to_lds %0, %1" :: "s"(g0), "s"(g1) : "memory");
}

// ---------------------------------------------------------------- sq norms
__global__ __launch_bounds__(256) void k_sqnorm(const float* __restrict__ x,
                                                float* __restrict__ sq) {
  int gid = blockIdx.x * 256 + threadIdx.x;        // over B*N
  int b = gid >> 11, n = gid & (N_ - 1);
  const float* xb = x + (size_t)b * C_ * N_ + n;   // x is [B][C][N]
  float s = 0.f;
#pragma unroll 8
  for (int c = 0; c < C_; ++c) { float v = xb[(size_t)c * N_]; s += v * v; }
  sq[gid] = s;
}

// ------------------------------------------------------- top-9 maintenance
static __device__ __forceinline__ void topk9_update(float d, int ci,
                                                    float bd[9], int bi[9],
                                                    float& wv, int& wp) {
  if (d < wv) {
#pragma unroll
    for (int j = 0; j < 9; ++j) if (j == wp) { bd[j] = d; bi[j] = ci; }
    wv = bd[0]; wp = 0;
#pragma unroll
    for (int j = 1; j < 9; ++j) if (bd[j] > wv) { wv = bd[j]; wp = j; }
  }
}

// ------------------------------------------------- kNN (WMMA + TDM double buffer)
// Single flat LDS array (only shared object -> base offset 0, so TDM lds_addr
// byte offsets below are exact).
#define LA_OFF  0        // 16 x 132 f32 : query tile [m][c]
#define LB0_OFF 2112     // 128 x 16 f32 : candidate tile buf0 [c][n] (TDM dest)
#define LB1_OFF 4160     // 128 x 16 f32 : candidate tile buf1
#define LD_OFF  6208     // 16 x 16 f32  : distance tile [M][N]
#define MD_OFF  6464     // 32 x 9 f32   : merge dists
#define MI_OFF  6752     // 32 x 9 i32   : merge indices
#define SMEM_F  7040

__global__ __launch_bounds__(32) void k_knn(const float* __restrict__ x,
                                            const float* __restrict__ sq,
                                            int* __restrict__ nn) {
  __shared__ float smem[SMEM_F];

  const int wg = blockIdx.x;
  const int b  = wg >> 7;
  const int n0 = (wg & 127) << 4;
  const int lane = threadIdx.x;
  const int ml = lane & 15;
  const int hi = lane >> 4;
  const int kb = hi * 2;
  const float* xb  = x  + (size_t)b * C_ * N_;
  const float* sqb = sq + b * N_;

  // kick off TDM for candidate tile 0 (overlaps with query staging below)
  tdm_load_tile16x128(LB0_OFF * 4u, xb);

  // stage query tile (coalesced across n)
  for (int i = lane; i < 16 * C_; i += 32) {
    int m = i & 15, c = i >> 4;
    smem[LA_OFF + m * 132 + c] = xb[(size_t)c * N_ + (n0 + m)];
  }
  __syncthreads();

  // A fragments: lane<16 holds K=0,1; lane>=16 holds K=2,3 (M = lane&15)
  v2f af[32];
#pragma unroll
  for (int ks = 0; ks < 32; ++ks) {
    af[ks].x = smem[LA_OFF + ml * 132 + 4 * ks + kb];
    af[ks].y = smem[LA_OFF + ml * 132 + 4 * ks + kb + 1];
  }
  float sqr[8];
#pragma unroll
  for (int v = 0; v < 8; ++v) sqr[v] = sqb[n0 + hi * 8 + v];

  float bd[9]; int bi9[9];
#pragma unroll
  for (int j = 0; j < 9; ++j) { bd[j] = 3.4e38f; bi9[j] = 0; }
  float wv = 3.4e38f; int wp = 0;

  for (int jt = 0; jt < N_ / 16; ++jt) {
    const int col0 = jt * 16;
    const float* lb = &smem[(jt & 1) ? LB1_OFF : LB0_OFF];

    if (jt + 1 < N_ / 16) {
      // WAR: make sure prior reads of the other buffer are complete, then
      // prefetch the next candidate tile into it while we compute.
      asm volatile("s_wait_dscnt 0x0" ::: "memory");
      tdm_load_tile16x128(((jt & 1) ? LB0_OFF : LB1_OFF) * 4u,
                          xb + (size_t)(col0 + 16));
      __builtin_amdgcn_s_wait_tensorcnt(1);   // current tile landed
    } else {
      __builtin_amdgcn_s_wait_tensorcnt(0);
    }

    v8f acc;
#pragma unroll
    for (int v = 0; v < 8; ++v) acc[v] = 0.f;
#pragma unroll
    for (int ks = 0; ks < 32; ++ks) {
      v2f bf;   // B frag: lb is [c][16]; lane<16: K=c..c+1, lane>=16: K=c+2..c+3
      bf.x = lb[(4 * ks + kb) * 16 + ml];
      bf.y = lb[(4 * ks + kb + 1) * 16 + ml];
      acc = wmma_f32_16x16x4(af[ks], bf, acc);
    }

    // dist = |a|^2 + |b|^2 - 2 a.b ; C layout: lane->N, vgpr->M (upper half +8)
    const float sqc = sqb[col0 + ml];
#pragma unroll
    for (int v = 0; v < 8; ++v)
      smem[LD_OFF + (hi * 8 + v) * 16 + ml] = sqr[v] + sqc - 2.f * acc[v];
    __syncthreads();

    const int r = lane >> 1, hh = lane & 1;
#pragma unroll
    for (int i = 0; i < 8; ++i) {
      float d = smem[LD_OFF + r * 16 + hh * 8 + i];
      topk9_update(d, col0 + hh * 8 + i, bd, bi9, wv, wp);
    }
    __syncthreads();
  }

  // merge the two half-row lanes, write 9 neighbor indices per row
  int* mi = (int*)smem;
#pragma unroll
  for (int j = 0; j < 9; ++j) {
    smem[MD_OFF + lane * 9 + j] = bd[j];
    mi[MI_OFF + lane * 9 + j] = bi9[j];
  }
  __syncthreads();
  if ((lane & 1) == 0) {
    const int r = lane >> 1;
#pragma unroll
    for (int j = 0; j < 9; ++j)
      topk9_update(smem[MD_OFF + (lane + 1) * 9 + j],
                   mi[MI_OFF + (lane + 1) * 9 + j], bd, bi9, wv, wp);
    int* dst = nn + (size_t)(b * N_ + n0 + r) * K_;
#pragma unroll
    for (int j = 0; j < 9; ++j) dst[j] = bi9[j];
  }
}

// ----------------------------------------------- pass 1: WMMA GEMM + stats
__global__ __launch_bounds__(128) void k_edge_stats(const float* __restrict__ x,
                                                    const float* __restrict__ Wm,
                                                    const int* __restrict__ nn,
                                                    float* __restrict__ gsum,
                                                    float* __restrict__ gsq) {
  __shared__ float feat[32][C2_ + 4];
  __shared__ float chS[COUT_];
  __shared__ float chQ[COUT_];

  const int wg = blockIdx.x;
  const int b  = wg >> 6;
  const int n0 = (wg & 63) << 5;
  const int tid = threadIdx.x;
  const int lane = tid & 31;
  const int w = tid >> 5;
  const int ml = lane & 15, hi = lane >> 4, kb = hi * 2;
  const float* xb = x + (size_t)b * C_ * N_;
  const int rT = tid & 31;          // feat row this thread fills (n-coalesced)
  const int nT = n0 + rT;

  float csum[4], csq[4];
#pragma unroll
  for (int t = 0; t < 4; ++t) { csum[t] = 0.f; csq[t] = 0.f; }

  for (int k = 0; k < K_; ++k) {
    const int idxT = nn[(b * N_ + nT) * K_ + k];
    for (int i = tid; i < 32 * C2_; i += 128) {
      int c = i >> 5;
      float val;
      if (c < C_) val = xb[(size_t)c * N_ + nT];
      else { int cc = c - C_; val = xb[(size_t)cc * N_ + idxT] - xb[(size_t)cc * N_ + nT]; }
      feat[rT][c] = val;
    }
    __syncthreads();

    v8f a0acc[4], a1acc[4];
#pragma unroll
    for (int t = 0; t < 4; ++t)
#pragma unroll
      for (int v = 0; v < 8; ++v) { a0acc[t][v] = 0.f; a1acc[t][v] = 0.f; }

    for (int ks = 0; ks < 64; ++ks) {
      v2f a0, a1;
      a0.x = feat[ml][4 * ks + kb];       a0.y = feat[ml][4 * ks + kb + 1];
      a1.x = feat[16 + ml][4 * ks + kb];  a1.y = feat[16 + ml][4 * ks + kb + 1];
#pragma unroll
      for (int t = 0; t < 4; ++t) {
        const int o = (w * 4 + t) * 16 + ml;       // B[k][n] = W[o=n][c=k]
        v2f bb;
        bb.x = Wm[(size_t)o * C2_ + 4 * ks + kb];
        bb.y = Wm[(size_t)o * C2_ + 4 * ks + kb + 1];
        a0acc[t] = wmma_f32_16x16x4(a0, bb, a0acc[t]);
        a1acc[t] = wmma_f32_16x16x4(a1, bb, a1acc[t]);
      }
    }
#pragma unroll
    for (int t = 0; t < 4; ++t)
#pragma unroll
      for (int v = 0; v < 8; ++v) {
        float h0 = a0acc[t][v], h1 = a1acc[t][v];
        csum[t] += h0 + h1;
        csq[t]  += h0 * h0 + h1 * h1;
      }
    __syncthreads();
  }

  for (int i = tid; i < COUT_; i += 128) { chS[i] = 0.f; chQ[i] = 0.f; }
  __syncthreads();
#pragma unroll
  for (int t = 0; t < 4; ++t) {
    const int ch = (w * 4 + t) * 16 + ml;
    atomicAdd(&chS[ch], csum[t]);
    atomicAdd(&chQ[ch], csq[t]);
  }
  __syncthreads();
  for (int i = tid; i < COUT_; i += 128) {
    atomicAdd(&gsum[i], chS[i]);
    atomicAdd(&gsq[i],  chQ[i]);
  }
}

// ------------------------------------------------------- BN stats finalize
__global__ __launch_bounds__(256) void k_finalize(const float* __restrict__ gsum,
                                                  const float* __restrict__ gsq,
                                                  const float* __restrict__ gamma,
                                                  const float* __restrict__ beta,
                                                  float* __restrict__ scl,
                                                  float* __restrict__ bia) {
  int o = threadIdx.x;
  const float invM = 1.0f / ((float)B_ * (float)N_ * (float)K_);
  float mean = gsum[o] * invM;
  float var  = gsq[o] * invM - mean * mean;
  float s = gamma[o] * rsqrtf(var + BN_EPS);
  scl[o] = s;
  bia[o] = beta[o] - mean * s;
}

// ------------------------------- pass 2: WMMA GEMM + BN + ReLU + max-over-K
__global__ __launch_bounds__(128) void k_edge_out(const float* __restrict__ x,
                                                  const float* __restrict__ Wm,
                                                  const int* __restrict__ nn,
                                                  const float* __restrict__ scl,
                                                  const float* __restrict__ bia,
                                                  float* __restrict__ out) {
  __shared__ float feat[32][C2_ + 4];   // reused as hout[32][256] at the end

  const int wg = blockIdx.x;
  const int b  = wg >> 6;
  const int n0 = (wg & 63) << 5;
  const int tid = threadIdx.x;
  const int lane = tid & 31;
  const int w = tid >> 5;
  const int ml = lane & 15, hi = lane >> 4, kb = hi * 2;
  const float* xb = x + (size_t)b * C_ * N_;
  const int rT = tid & 31;
  const int nT = n0 + rT;

  float sc[4], bs[4];
#pragma unroll
  for (int t = 0; t < 4; ++t) {
    const int o = (w * 4 + t) * 16 + ml;
    sc[t] = scl[o]; bs[t] = bia[o];
  }
  float m0[4][8], m1[4][8];
#pragma unroll
  for (int t = 0; t < 4; ++t)
#pragma unroll
    for (int v = 0; v < 8; ++v) { m0[t][v] = 0.f; m1[t][v] = 0.f; }

  for (int k = 0; k < K_; ++k) {
    const int idxT = nn[(b * N_ + nT) * K_ + k];
    for (int i = tid; i < 32 * C2_; i += 128) {
      int c = i >> 5;
      float val;
      if (c < C_) val = xb[(size_t)c * N_ + nT];
      else { int cc = c - C_; val = xb[(size_t)cc * N_ + idxT] - xb[(size_t)cc * N_ + nT]; }
      feat[rT][c] = val;
    }
    __syncthreads();

    v8f a0acc[4], a1acc[4];
#pragma unroll
    for (int t = 0; t < 4; ++t)
#pragma unroll
      for (int v = 0; v < 8; ++v) { a0acc[t][v] = 0.f; a1acc[t][v] = 0.f; }

    for (int ks = 0; ks < 64; ++ks) {
      v2f a0, a1;
      a0.x = feat[ml][4 * ks + kb];       a0.y = feat[ml][4 * ks + kb + 1];
      a1.x = feat[16 + ml][4 * ks + kb];  a1.y = feat[16 + ml][4 * ks + kb + 1];
#pragma unroll
      for (int t = 0; t < 4; ++t) {
        const int o = (w * 4 + t) * 16 + ml;
        v2f bb;
        bb.x = Wm[(size_t)o * C2_ + 4 * ks + kb];
        bb.y = Wm[(size_t)o * C2_ + 4 * ks + kb + 1];
        a0acc[t] = wmma_f32_16x16x4(a0, bb, a0acc[t]);
        a1acc[t] = wmma_f32_16x16x4(a1, bb, a1acc[t]);
      }
    }
#pragma unroll
    for (int t = 0; t < 4; ++t)
#pragma unroll
      for (int v = 0; v < 8; ++v) {
        float h0 = fmaxf(fmaf(a0acc[t][v], sc[t], bs[t]), 0.f);
        float h1 = fmaxf(fmaf(a1acc[t][v], sc[t], bs[t]), 0.f);
        m0[t][v] = fmaxf(m0[t][v], h0);
        m1[t][v] = fmaxf(m1[t][v], h1);
      }
    __syncthreads();
  }

  // transpose through LDS for coalesced stores: out[b][o][n]
  float* hout = &feat[0][0];
#pragma unroll
  for (int t = 0; t < 4; ++t) {
    const int ch = (w * 4 + t) * 16 + ml;
#pragma unroll
    for (int v = 0; v < 8; ++v) {
      hout[(hi * 8 + v) * C2_ + ch]        = m0[t][v];
      hout[(16 + hi * 8 + v) * C2_ + ch]   = m1[t][v];
    }
  }
  __syncthreads();
  float* ob = out + (size_t)b * COUT_ * N_;
  for (int i = tid; i < 32 * COUT_; i += 128) {
    int o = i >> 5;
    ob[(size_t)o * N_ + n0 + rT] = hout[rT * C2_ + o];
  }
}

// ------------------------------------------------------------------ launch
extern "C" void kernel_launch(void* const* d_in, const int* in_sizes, int n_in,
                              void* d_out, int out_size, void* d_ws, size_t ws_size,
                              hipStream_t stream) {
  const float* x     = (const float*)d_in[0];   // (8,128,2048,1) -> [B][C][N]
  const float* Wm    = (const float*)d_in[1];   // (256, 256)
  const float* gamma = (const float*)d_in[2];
  const float* beta  = (const float*)d_in[3];
  float* out = (float*)d_out;                   // (8, 256, 2048)

  char* ws = (char*)d_ws;
  float* sq   = (float*)ws;                           // 16384 floats
  int*   nn   = (int*)(ws + 65536);                   // 147456 ints
  float* gsum = (float*)(ws + 65536 + 589824);        // 256
  float* gsq  = gsum + 256;
  float* scl  = gsum + 512;
  float* bia  = gsum + 768;

  k_sqnorm<<<(B_ * N_) / 256, 256, 0, stream>>>(x, sq);
  k_knn<<<B_ * (N_ / 16), 32, 0, stream>>>(x, sq, nn);
  hipMemsetAsync(gsum, 0, 2 * COUT_ * sizeof(float), stream);
  k_edge_stats<<<B_ * (N_ / 32), 128, 0, stream>>>(x, Wm, nn, gsum, gsq);
  k_finalize<<<1, 256, 0, stream>>>(gsum, gsq, gamma, beta, scl, bia);
  k_edge_out<<<B_ * (N_ / 32), 128, 0, stream>>>(x, Wm, nn, scl, bia, out);
}